// HLDeconfounder_12197707120841
// MI455X (gfx1250) — compile-verified
//
#include <hip/hip_runtime.h>
#include <hip/hip_bf16.h>

#define N_NODE 10000
#define E_EDGE 320000
#define E_LNK  1280000
#define DD     128
#define EF     24
#define SEQ    64

typedef __attribute__((ext_vector_type(2))) float v2f;
typedef __attribute__((ext_vector_type(8))) float v8f;

// ---------------------------------------------------------------------------
// 1) Wc[k] = W_hl[k] @ W_ec  (padded [3][128][16]), bc = b_hl@W_ec + b_ec (pad 16)
// ---------------------------------------------------------------------------
__global__ void wc_kernel(const float* __restrict__ W_hl, const float* __restrict__ W_ec,
                          const float* __restrict__ b_hl, const float* __restrict__ b_ec,
                          float* __restrict__ Wc, float* __restrict__ bc) {
    if (blockIdx.x < 24) {
        int gid = blockIdx.x * 256 + threadIdx.x;          // [0, 6144)
        int c = gid & 15, row = (gid >> 4) & 127, k = gid >> 11;
        float s = 0.f;
        if (c < 6) {
            const float* wh = W_hl + (size_t)k * DD * DD + (size_t)row * DD;
            for (int m = 0; m < DD; ++m) s += wh[m] * W_ec[m * 6 + c];
        }
        Wc[gid] = s;
    } else {
        int c = threadIdx.x;
        if (c < 16) {
            float s = 0.f;
            if (c < 6) {
                s = b_ec[c];
                for (int m = 0; m < DD; ++m) s += b_hl[m] * W_ec[m * 6 + c];
            }
            bc[c] = s;
        }
    }
}

// ---------------------------------------------------------------------------
// 2) Tx0 = x_link @ W_start + b_start      [E,24]@[24,128]  (fp32 WMMA, K=24)
//    block: 256 thr = 8 waves; 16 edge-rows x 128 cols; grid = E/16
// ---------------------------------------------------------------------------
__global__ void start_kernel(const float* __restrict__ x_link, const float* __restrict__ W_start,
                             const float* __restrict__ b_start, float* __restrict__ Tx0) {
    __shared__ float sW[EF * DD];    // [kk][d]  12 KB
    __shared__ float sX[16 * EF];    // [row][kk] 1.5 KB
    int tid = threadIdx.x;
    for (int i = tid; i < EF * DD; i += 256) sW[i] = W_start[i];
    int row0 = blockIdx.x * 16;
    for (int i = tid; i < 16 * EF; i += 256) sX[i] = x_link[(size_t)row0 * EF + i];
    __syncthreads();

    int wv = tid >> 5, lane = tid & 31, lo = lane & 15, hi = lane >> 4;
    v8f acc = {};
    for (int k0 = 0; k0 < EF; k0 += 4) {
        v2f a, b;
        a.x = sX[lo * EF + k0 + 2 * hi];
        a.y = sX[lo * EF + k0 + 2 * hi + 1];
        b.x = sW[(k0 + 2 * hi) * DD + wv * 16 + lo];
        b.y = sW[(k0 + 2 * hi + 1) * DD + wv * 16 + lo];
        acc = __builtin_amdgcn_wmma_f32_16x16x4_f32(false, a, false, b, (short)0, acc, false, false);
    }
    float bias = b_start[wv * 16 + lo];
    for (int j = 0; j < 8; ++j) {
        size_t r = (size_t)(row0 + 8 * hi + j);
        Tx0[r * DD + wv * 16 + lo] = acc[j] + bias;
    }
}

// ---------------------------------------------------------------------------
// 3) h_ent[n,d] = sum_t h_entity[t,n,d]*W_t[t] + b_t   (memory-bound stream)
// ---------------------------------------------------------------------------
__global__ void temporal_kernel(const float* __restrict__ h_entity, const float* __restrict__ W_t,
                                const float* __restrict__ b_t, float* __restrict__ h_ent) {
    __shared__ float sw[SEQ];
    if (threadIdx.x < SEQ) sw[threadIdx.x] = W_t[threadIdx.x];
    __syncthreads();
    size_t gid = (size_t)blockIdx.x * 256 + threadIdx.x;   // float4 index, N*D/4 total
    const float4* src = (const float4*)h_entity;
    float4 acc = {0.f, 0.f, 0.f, 0.f};
    for (int t = 0; t < SEQ; ++t) {
        float4 v = src[(size_t)t * (N_NODE * DD / 4) + gid];
        float w = sw[t];
        acc.x = fmaf(v.x, w, acc.x); acc.y = fmaf(v.y, w, acc.y);
        acc.z = fmaf(v.z, w, acc.z); acc.w = fmaf(v.w, w, acc.w);
    }
    float b = b_t[0];
    acc.x += b; acc.y += b; acc.z += b; acc.w += b;
    ((float4*)h_ent)[gid] = acc;
}

// ---------------------------------------------------------------------------
// 4) o = a*x + b*y   (float4 stream; used for Laguerre recurrence init)
// ---------------------------------------------------------------------------
__global__ void axpy_kernel(float* __restrict__ o, const float* __restrict__ x,
                            const float* __restrict__ y, float a, float b) {
    size_t gid = (size_t)blockIdx.x * 256 + threadIdx.x;
    float4 xv = ((const float4*)x)[gid];
    float4 yv = ((const float4*)y)[gid];
    float4 r = {a * xv.x + b * yv.x, a * xv.y + b * yv.y,
                a * xv.z + b * yv.z, a * xv.w + b * yv.w};
    ((float4*)o)[gid] = r;
}

__global__ void zero_kernel(float* __restrict__ o) {
    size_t gid = (size_t)blockIdx.x * 256 + threadIdx.x;
    float4 z = {0.f, 0.f, 0.f, 0.f};
    ((float4*)o)[gid] = z;
}

// ---------------------------------------------------------------------------
// 5) y[dst[e]] += coef * ew[e*ws+wo] * x[src[e]]   (one wave per edge, D=128,
//    float4 gather + 4x global_atomic_add_f32 scatter per lane)
// ---------------------------------------------------------------------------
__global__ void prop_kernel(const float* __restrict__ x, float* __restrict__ y,
                            const int* __restrict__ src, const int* __restrict__ dst,
                            const float* __restrict__ ew, int wstride, int woff, float coef) {
    int e = blockIdx.x * 8 + (threadIdx.x >> 5);
    int lane = threadIdx.x & 31;
    int s = src[e], d = dst[e];
    float w = ew[(size_t)e * wstride + woff] * coef;
    float4 v = ((const float4*)(x + (size_t)s * DD))[lane];
    float* yp = y + (size_t)d * DD + lane * 4;
    atomicAdd(yp + 0, v.x * w);
    atomicAdd(yp + 1, v.y * w);
    atomicAdd(yp + 2, v.z * w);
    atomicAdd(yp + 3, v.w * w);
}

// ---------------------------------------------------------------------------
// 6) score = Tx0@Wc0 + Tx1@Wc1 + Tx2@Wc2 + bc ; wk[e,k] = 1/(1+exp(s1-s0))
//    block: 128 thr = 4 waves, 64 edge-rows; grid = E/64. fp32 WMMA K=128 x3.
//    LDS: 64x128 A tile (32 KB) + all Wc (24 KB)
// ---------------------------------------------------------------------------
__global__ void score_kernel(const float* __restrict__ Tx0, const float* __restrict__ Tx1,
                             const float* __restrict__ Tx2, const float* __restrict__ Wc,
                             const float* __restrict__ bc, float* __restrict__ wk) {
    __shared__ float sA[64 * DD];        // 32 KB
    __shared__ float sB[3 * DD * 16];    // 24 KB
    int tid = threadIdx.x;
    for (int i = tid; i < 3 * DD * 16; i += 128) sB[i] = Wc[i];

    int row0 = blockIdx.x * 64;
    int wv = tid >> 5, lane = tid & 31, lo = lane & 15, hi = lane >> 4;
    v8f acc = {};
    const float* Txs[3] = {Tx0, Tx1, Tx2};
    for (int k = 0; k < 3; ++k) {
        __syncthreads();
        const float4* g = (const float4*)(Txs[k] + (size_t)row0 * DD);
        float4* s4 = (float4*)sA;
        for (int i = tid; i < 64 * DD / 4; i += 128) s4[i] = g[i];
        __syncthreads();
        const float* Bk = sB + k * DD * 16;
        for (int k0 = 0; k0 < DD; k0 += 4) {
            v2f a, b;
            a.x = sA[(wv * 16 + lo) * DD + k0 + 2 * hi];
            a.y = sA[(wv * 16 + lo) * DD + k0 + 2 * hi + 1];
            b.x = Bk[(k0 + 2 * hi) * 16 + lo];
            b.y = Bk[(k0 + 2 * hi + 1) * 16 + lo];
            acc = __builtin_amdgcn_wmma_f32_16x16x4_f32(false, a, false, b, (short)0, acc, false, false);
        }
    }
    float bcv = bc[lo];
    for (int j = 0; j < 8; ++j) {
        float sc = acc[j] + bcv;   // score[row0+wv*16+8*hi+j][lo]
        for (int k = 0; k < 3; ++k) {
            float s0 = __shfl(sc, 2 * k + 16 * hi, 32);
            float s1 = __shfl(sc, 2 * k + 1 + 16 * hi, 32);
            if (lo == k) {
                int row = row0 + wv * 16 + 8 * hi + j;
                wk[(size_t)row * 3 + k] = 1.0f / (1.0f + __expf(s1 - s0));
            }
        }
    }
}

// ---------------------------------------------------------------------------
// 7) out[n,d] = b_gcn[d]  (accumulator init)
// ---------------------------------------------------------------------------
__global__ void out_init_kernel(float* __restrict__ out, const float* __restrict__ b_gcn) {
    size_t gid = (size_t)blockIdx.x * 256 + threadIdx.x;   // float4 index
    int c = (int)((gid * 4) & (DD - 1));
    float4 r = {b_gcn[c], b_gcn[c + 1], b_gcn[c + 2], b_gcn[c + 3]};
    ((float4*)out)[gid] = r;
}

// ---------------------------------------------------------------------------
// 8) out += Hin @ Wg    [10000,128]@[128,128]  fp32 WMMA, K=128
//    block: 256 thr = 8 waves; 16 rows x 128 cols; grid = 625
//    LDS: 16x128 A (8 KB) + 64x128 B half (32 KB), two K halves
// ---------------------------------------------------------------------------
__global__ void gemm_acc_kernel(const float* __restrict__ Hin, const float* __restrict__ Wg,
                                float* __restrict__ out) {
    __shared__ float sA[16 * DD];   // 8 KB
    __shared__ float sB[64 * DD];   // 32 KB
    int tid = threadIdx.x, wv = tid >> 5, lane = tid & 31, lo = lane & 15, hi = lane >> 4;
    int row0 = blockIdx.x * 16;

    const float4* ga = (const float4*)(Hin + (size_t)row0 * DD);
    float4* s4a = (float4*)sA;
    for (int i = tid; i < 16 * DD / 4; i += 256) s4a[i] = ga[i];

    v8f acc;
    for (int j = 0; j < 8; ++j)
        acc[j] = out[(size_t)(row0 + 8 * hi + j) * DD + wv * 16 + lo];

    for (int h = 0; h < 2; ++h) {
        __syncthreads();
        const float4* gb = (const float4*)(Wg + (size_t)h * 64 * DD);
        float4* s4b = (float4*)sB;
        for (int i = tid; i < 64 * DD / 4; i += 256) s4b[i] = gb[i];
        __syncthreads();
        for (int k0 = 0; k0 < 64; k0 += 4) {
            v2f a, b;
            a.x = sA[lo * DD + h * 64 + k0 + 2 * hi];
            a.y = sA[lo * DD + h * 64 + k0 + 2 * hi + 1];
            b.x = sB[(k0 + 2 * hi) * DD + wv * 16 + lo];
            b.y = sB[(k0 + 2 * hi + 1) * DD + wv * 16 + lo];
            acc = __builtin_amdgcn_wmma_f32_16x16x4_f32(false, a, false, b, (short)0, acc, false, false);
        }
    }
    for (int j = 0; j < 8; ++j)
        out[(size_t)(row0 + 8 * hi + j) * DD + wv * 16 + lo] = acc[j];
}

// ---------------------------------------------------------------------------
extern "C" void kernel_launch(void* const* d_in, const int* in_sizes, int n_in,
                              void* d_out, int out_size, void* d_ws, size_t ws_size,
                              hipStream_t stream) {
    const float* h_entity = (const float*)d_in[0];
    const float* x_link   = (const float*)d_in[1];
    const float* ewl      = (const float*)d_in[2];
    const int*   eil      = (const int*)d_in[3];   // [2, E_LNK]
    const int*   ei       = (const int*)d_in[4];   // [2, E]
    const float* W_start  = (const float*)d_in[5];
    const float* b_start  = (const float*)d_in[6];
    const float* W_hl     = (const float*)d_in[7];
    const float* b_hl     = (const float*)d_in[8];
    const float* W_ec     = (const float*)d_in[9];
    const float* b_ec     = (const float*)d_in[10];
    const float* W_t      = (const float*)d_in[11];
    const float* b_t      = (const float*)d_in[12];
    const float* W_gcn    = (const float*)d_in[13];
    const float* b_gcn    = (const float*)d_in[14];
    float* out = (float*)d_out;

    // workspace layout (floats); total ~487 MiB
    float* ws   = (float*)d_ws;
    float* Tx0  = ws;
    float* Tx1  = Tx0 + (size_t)E_EDGE * DD;
    float* Tx2  = Tx1 + (size_t)E_EDGE * DD;
    float* Wc   = Tx2 + (size_t)E_EDGE * DD;
    float* bc   = Wc + 3 * DD * 16;
    float* wk   = bc + 16;
    float* hent = wk + (size_t)E_EDGE * 3;
    float* hA   = hent + (size_t)N_NODE * DD;
    float* hB   = hA + (size_t)N_NODE * DD;

    const int TXB = (E_EDGE * DD) / 1024;     // 40000 blocks of 256 thr x float4
    const int NDB = (N_NODE * DD) / 1024;     // 1250

    // fold W_hl/W_ec into Wc, bc
    wc_kernel<<<25, 256, 0, stream>>>(W_hl, W_ec, b_hl, b_ec, Wc, bc);
    // Tx0 = start MLP (WMMA)
    start_kernel<<<E_EDGE / 16, 256, 0, stream>>>(x_link, W_start, b_start, Tx0);
    // temporal projection
    temporal_kernel<<<NDB, 256, 0, stream>>>(h_entity, W_t, b_t, hent);
    // Tx1 = Tx0 - L Tx0
    axpy_kernel<<<TXB, 256, 0, stream>>>(Tx1, Tx0, Tx0, 1.0f, 0.0f);
    prop_kernel<<<E_LNK / 8, 256, 0, stream>>>(Tx0, Tx1, eil, eil + E_LNK, ewl, 1, 0, -1.0f);
    // Tx2 = 1.5 Tx1 - 0.5 Tx0 - 0.5 L Tx1
    axpy_kernel<<<TXB, 256, 0, stream>>>(Tx2, Tx1, Tx0, 1.5f, -0.5f);
    prop_kernel<<<E_LNK / 8, 256, 0, stream>>>(Tx1, Tx2, eil, eil + E_LNK, ewl, 1, 0, -0.5f);
    // fused score + softmax -> per-hop edge weights wk[E,3] (WMMA)
    score_kernel<<<E_EDGE / 64, 128, 0, stream>>>(Tx0, Tx1, Tx2, Wc, bc, wk);
    // out = b_gcn
    out_init_kernel<<<NDB, 256, 0, stream>>>(out, b_gcn);
    // 3-hop GCN: propagate then accumulate GEMM (WMMA)
    float* hcur = hent;
    float* hbuf[2] = {hA, hB};
    for (int k = 0; k < 3; ++k) {
        float* hnext = hbuf[k & 1];
        zero_kernel<<<NDB, 256, 0, stream>>>(hnext);
        prop_kernel<<<E_EDGE / 8, 256, 0, stream>>>(hcur, hnext, ei, ei + E_EDGE, wk, 3, k, 1.0f);
        gemm_acc_kernel<<<N_NODE / 16, 256, 0, stream>>>(hnext, W_gcn + (size_t)k * DD * DD, out);
        hcur = hnext;
    }
    (void)in_sizes; (void)n_in; (void)out_size; (void)ws_size;
}